// DynamicLabelAssignment_22522808500280
// MI455X (gfx1250) — compile-verified
//
#include <hip/hip_runtime.h>
#include <math.h>

#define NCLS 80
#define TOPK 10
#define RADIUS_F 2.5f

typedef float v2f __attribute__((ext_vector_type(2)));
typedef float v8f __attribute__((ext_vector_type(8)));

__device__ __forceinline__ float softplusf(float x) {
    // log(1 + e^x) = max(x,0) + log1p(exp(-|x|))
    return fmaxf(x, 0.0f) + log1pf(expf(-fabsf(x)));
}

__device__ __forceinline__ bool keyless(float c0, int i0, float c1, int i1) {
    return (c0 < c1) || (c0 == c1 && i0 < i1);
}

// -------------------------------------------------------------------------
// Kernel 1: total_neg[a] = sum_c softplus(pred_scores[a,c]) via WMMA f32 16x16x4
// Each wave (32 lanes) owns 16 anchor rows. A-matrix tile = softplus values,
// B = all-ones => D[m,n] = row sum for anchor m (all columns identical).
// A layout (16x4 f32): lanes 0-15 M=0..15 K={0,1}; lanes 16-31 K={2,3}.
// D layout: lane 0 VGPR i -> M=i; lane 16 VGPR i -> M=8+i.
// -------------------------------------------------------------------------
__global__ __launch_bounds__(256)
void k_rowsum_wmma(const float* __restrict__ ps, float* __restrict__ tneg, int A) {
    const int wave = threadIdx.x >> 5;
    const int lane = threadIdx.x & 31;
    const int base = blockIdx.x * 128 + wave * 16;
    const int m    = base + (lane & 15);
    const int kb   = (lane >> 4) << 1;      // 0 or 2
    const bool valid = (m < A);

    const float2* row2 = (const float2*)(ps + (long)m * NCLS);

    v8f c = {};
    v2f b; b[0] = 1.0f; b[1] = 1.0f;

    #pragma unroll
    for (int c0 = 0; c0 < NCLS; c0 += 4) {
        float x0 = 0.0f, x1 = 0.0f;
        if (valid) {
            float2 v = row2[(c0 + kb) >> 1];
            x0 = v.x; x1 = v.y;
        }
        v2f a; a[0] = softplusf(x0); a[1] = softplusf(x1);
        c = __builtin_amdgcn_wmma_f32_16x16x4_f32(
                /*neg_a=*/false, a, /*neg_b=*/false, b,
                /*c_mod=*/(short)0, c, /*reuse_a=*/false, /*reuse_b=*/false);
    }

    if ((lane & 15) == 0) {                 // lanes 0 and 16
        const int mb = base + (lane >> 4) * 8;
        #pragma unroll
        for (int i = 0; i < 8; ++i)
            if (mb + i < A) tneg[mb + i] = c[i];
    }
}

// -------------------------------------------------------------------------
// Kernel 2: one block per GT. Register top-10 per thread + LDS merge.
// -------------------------------------------------------------------------
__global__ __launch_bounds__(256)
void k_pergt(const float* __restrict__ ps, const float* __restrict__ pb,
             const float* __restrict__ ap, const int* __restrict__ glab,
             const float* __restrict__ gbb, const float* __restrict__ tneg,
             int* __restrict__ topk, int* __restrict__ dynk, int A) {
    const int g = blockIdx.x;
    const int t = threadIdx.x;

    const float gx1 = gbb[g * 4 + 0], gy1 = gbb[g * 4 + 1];
    const float gx2 = gbb[g * 4 + 2], gy2 = gbb[g * 4 + 3];
    const float cx = 0.5f * (gx1 + gx2), cy = 0.5f * (gy1 + gy2);
    const float rx = RADIUS_F * (gx2 - gx1), ry = RADIUS_F * (gy2 - gy1);
    const float area_g = (gx2 - gx1) * (gy2 - gy1);
    const int label = glab[g];
    const float* psL = ps + label;

    float tc[TOPK]; int ti[TOPK];
    #pragma unroll
    for (int j = 0; j < TOPK; ++j) { tc[j] = INFINITY; ti[j] = 0x7fffffff; }
    int cnt = 0;

    for (int a = t; a < A; a += 256) {
        const float4 p  = ((const float4*)pb)[a];
        const float2 an = ((const float2*)ap)[a];

        const bool in_gt = (an.x >= gx1) & (an.x <= gx2) & (an.y >= gy1) & (an.y <= gy2);
        const bool in_c  = (an.x >= cx - rx) & (an.x <= cx + rx) &
                           (an.y >= cy - ry) & (an.y <= cy + ry);
        const bool inside = in_gt & in_c;

        const float lx = fmaxf(p.x, gx1), ly = fmaxf(p.y, gy1);
        const float hx = fminf(p.z, gx2), hy = fminf(p.w, gy2);
        const float ov = fmaxf(hx - lx, 0.0f) * fmaxf(hy - ly, 0.0f);
        const float area_p = (p.z - p.x) * (p.w - p.y);
        const float iou = ov / (area_p + area_g - ov + 1e-6f);

        // cls_cost = total_neg - score[a,label]  (softplus(-s)-softplus(s) == -s)
        float cost = (tneg[a] - psL[(long)a * NCLS])
                   - 3.0f * logf(fmaxf(iou, 1e-7f));
        if (!inside) cost += 1e10f;

        cnt += (inside && (iou > 0.0f)) ? 1 : 0;

        if (keyless(cost, a, tc[TOPK - 1], ti[TOPK - 1])) {
            float cc = cost; int ca = a;
            #pragma unroll
            for (int j = 0; j < TOPK; ++j) {
                const bool l = keyless(cc, ca, tc[j], ti[j]);
                const float oc = tc[j]; const int oi = ti[j];
                if (l) { tc[j] = cc; ti[j] = ca; cc = oc; ca = oi; }
            }
        }
    }

    __shared__ float lc[256 * TOPK];
    __shared__ int   li[256 * TOPK];
    __shared__ float rc[256];
    __shared__ int   ri[256];
    __shared__ int   scnt;

    if (t == 0) scnt = 0;
    #pragma unroll
    for (int j = 0; j < TOPK; ++j) { lc[t * TOPK + j] = tc[j]; li[t * TOPK + j] = ti[j]; }
    __syncthreads();
    atomicAdd(&scnt, cnt);

    int ptr = 0;
    for (int r = 0; r < TOPK; ++r) {
        rc[t] = (ptr < TOPK) ? lc[t * TOPK + ptr] : INFINITY;
        ri[t] = (ptr < TOPK) ? li[t * TOPK + ptr] : 0x7fffffff;
        __syncthreads();
        for (int s = 128; s > 0; s >>= 1) {
            if (t < s) {
                if (keyless(rc[t + s], ri[t + s], rc[t], ri[t])) {
                    rc[t] = rc[t + s]; ri[t] = ri[t + s];
                }
            }
            __syncthreads();
        }
        const int win = ri[0];
        if (t == 0) topk[g * TOPK + r] = win;
        if ((win % 256) == t) ptr++;     // owner advances (a == t mod 256)
        __syncthreads();
    }

    if (t == 0) {
        int k = scnt;
        if (k < 1) k = 1;
        if (k > TOPK) k = TOPK;
        dynk[g] = k;
    }
}

// -------------------------------------------------------------------------
// Kernel 3a: zero scores region of d_out, init assigned = -1
// -------------------------------------------------------------------------
__global__ void k_init(float* __restrict__ scores, int* __restrict__ assigned,
                       long nsc, int A) {
    const long i = (long)blockIdx.x * blockDim.x + threadIdx.x;
    if (i < nsc) scores[i] = 0.0f;
    if (i < A) assigned[i] = -1;
}

// -------------------------------------------------------------------------
// Kernel 3b: scatter-max assignment (matches jax .at[].max semantics)
// -------------------------------------------------------------------------
__global__ void k_assign(const int* __restrict__ topk, const int* __restrict__ dynk,
                         int* __restrict__ assigned, int G, int A) {
    const int i = blockIdx.x * blockDim.x + threadIdx.x;
    if (i >= G * TOPK) return;
    const int g = i / TOPK, r = i % TOPK;
    if (r < dynk[g]) {
        const int a = topk[i];
        if (a >= 0 && a < A) atomicMax(&assigned[a], g);
    }
}

// -------------------------------------------------------------------------
// Kernel 4: finalize labels / bboxes / scores per anchor
// -------------------------------------------------------------------------
__global__ __launch_bounds__(256)
void k_final(const float* __restrict__ pb, const int* __restrict__ glab,
             const float* __restrict__ gbb, const int* __restrict__ assigned,
             float* __restrict__ out_labels, float* __restrict__ out_bb,
             float* __restrict__ out_sc, int A) {
    const int a = blockIdx.x * blockDim.x + threadIdx.x;
    if (a >= A) return;

    const int g = assigned[a];
    const bool pos = (g >= 0);
    const int safe = pos ? g : 0;
    const int label = pos ? glab[safe] : NCLS;

    out_labels[a] = (float)label;

    const float4 gb = ((const float4*)gbb)[safe];
    out_bb[(long)a * 4 + 0] = pos ? gb.x : 0.0f;
    out_bb[(long)a * 4 + 1] = pos ? gb.y : 0.0f;
    out_bb[(long)a * 4 + 2] = pos ? gb.z : 0.0f;
    out_bb[(long)a * 4 + 3] = pos ? gb.w : 0.0f;

    const float4 p = ((const float4*)pb)[a];
    const float lx = fmaxf(p.x, gb.x), ly = fmaxf(p.y, gb.y);
    const float hx = fminf(p.z, gb.z), hy = fminf(p.w, gb.w);
    const float ov = fmaxf(hx - lx, 0.0f) * fmaxf(hy - ly, 0.0f);
    const float area_p = (p.z - p.x) * (p.w - p.y);
    const float area_g = (gb.z - gb.x) * (gb.w - gb.y);
    const float iou = ov / (area_p + area_g - ov + 1e-6f);

    out_sc[(long)a * (NCLS + 1) + label] = pos ? iou : 0.0f;
}

// -------------------------------------------------------------------------
// launch
// -------------------------------------------------------------------------
extern "C" void kernel_launch(void* const* d_in, const int* in_sizes, int n_in,
                              void* d_out, int out_size, void* d_ws, size_t ws_size,
                              hipStream_t stream) {
    const float* ps   = (const float*)d_in[0];   // pred_scores   (A,80)
    const float* pb   = (const float*)d_in[1];   // pred_bboxes   (A,4)
    const float* ap   = (const float*)d_in[2];   // anchor_points (A,2)
    const int*   glab = (const int*)d_in[3];     // gt_labels     (G,)
    const float* gbb  = (const float*)d_in[4];   // gt_bboxes     (G,4)

    const int A = in_sizes[0] / NCLS;
    const int G = in_sizes[3];

    // workspace layout
    float* tneg     = (float*)d_ws;                        // A floats
    int*   assigned = (int*)((char*)d_ws + (size_t)A * 4); // A ints
    int*   topk     = assigned + A;                        // G*TOPK ints
    int*   dynk     = topk + G * TOPK;                     // G ints

    // output layout: labels [0,A) | bboxes [A,5A) | scores [5A, 86A)
    float* out        = (float*)d_out;
    float* out_labels = out;
    float* out_bb     = out + (long)A;
    float* out_sc     = out + (long)A * 5;

    k_rowsum_wmma<<<(A + 127) / 128, 256, 0, stream>>>(ps, tneg, A);

    const long nsc = (long)A * (NCLS + 1);
    k_init<<<(int)((nsc + 255) / 256), 256, 0, stream>>>(out_sc, assigned, nsc, A);

    k_pergt<<<G, 256, 0, stream>>>(ps, pb, ap, glab, gbb, tneg, topk, dynk, A);

    k_assign<<<(G * TOPK + 255) / 256, 256, 0, stream>>>(topk, dynk, assigned, G, A);

    k_final<<<(A + 255) / 256, 256, 0, stream>>>(pb, glab, gbb, assigned,
                                                 out_labels, out_bb, out_sc, A);
}